// DTWLoss_6459630813838
// MI455X (gfx1250) — compile-verified
//
#include <hip/hip_runtime.h>

#define DTW_N    1024
#define DTW_B    32
#define DTW_W    20
#define DTW_BAND 41
#define DTW_K    (2 * DTW_N - 1)
#define BIGF     1e30f

// ---------------------------------------------------------------------------
// CDNA5 async global->LDS staging (guarded: falls back to plain copies)
// ---------------------------------------------------------------------------
#if __has_builtin(__builtin_amdgcn_global_load_async_to_lds_b128) && \
    __has_builtin(__builtin_amdgcn_s_wait_asynccnt)
#define DTW_ASYNC 1
#endif

typedef int v4i_t __attribute__((ext_vector_type(4)));
typedef __attribute__((address_space(1))) v4i_t* glb_v4i;
typedef __attribute__((address_space(3))) v4i_t* lds_v4i;

__device__ __forceinline__ lds_v4i to_lds(const void* p) {
  // generic LDS pointer: low 32 bits are the LDS byte offset (ISA aperture rule)
  return (lds_v4i)(unsigned long long)p;
}
__device__ __forceinline__ glb_v4i to_glb(const void* p) {
  return (glb_v4i)(unsigned long long)p;
}

__device__ __forceinline__ float softplusf(float x) {
  // jax.nn.softplus: max(x,0) + log1p(exp(-|x|))
  return fmaxf(x, 0.0f) + log1pf(expf(-fabsf(x)));
}

// ---------------------------------------------------------------------------
// Kernel 1: one wave32 per batch element. Banded DTW DP over anti-diagonals
// (band width <= 21 == fits one wave32), choices recorded in LDS, then a
// sequential backtrack + loss accumulation by lane 0.
// ---------------------------------------------------------------------------
__global__ void __launch_bounds__(32)
dtw_kernel(const float* __restrict__ preds, const float* __restrict__ targs,
           const float* __restrict__ subcoef, float* __restrict__ part)
{
  extern __shared__ char smem[];
  float4*        s_pred = (float4*)(smem);                         // 16 KB
  float4*        s_targ = (float4*)(smem + 16 * DTW_N);            // 16 KB
  float*         s_D    = (float*)(smem + 32 * DTW_N);             // 3*N f32 = 12 KB
  unsigned char* s_ch   = (unsigned char*)(smem + 44 * DTW_N);     // N*41  = 41 KB

  const int b    = blockIdx.x;
  const int lane = threadIdx.x;

  const float4* pg = (const float4*)(preds + (size_t)b * DTW_N * 4);
  const float4* tg = (const float4*)(targs + (size_t)b * DTW_N * 4);

  __builtin_prefetch(pg, 0, 3);   // -> global_prefetch_b8
  __builtin_prefetch(tg, 0, 3);

  // ---- Phase A: stage sequences into LDS -------------------------------
#ifdef DTW_ASYNC
  for (int n = lane; n < DTW_N; n += 32) {
    __builtin_amdgcn_global_load_async_to_lds_b128(to_glb(pg + n), to_lds(s_pred + n), 0, 0);
    __builtin_amdgcn_global_load_async_to_lds_b128(to_glb(tg + n), to_lds(s_targ + n), 0, 0);
  }
  __builtin_amdgcn_s_wait_asynccnt(0);
#else
  for (int n = lane; n < DTW_N; n += 32) {
    s_pred[n] = pg[n];
    s_targ[n] = tg[n];
  }
#endif
  // ---- Phase B: init wavefront buffers to BIG --------------------------
  for (int n = lane; n < 3 * DTW_N; n += 32) s_D[n] = BIGF;
  __syncthreads();

  // ---- Phase C: anti-diagonal DP ---------------------------------------
  for (int k = 0; k < DTW_K; ++k) {
    float* W   = s_D + (k % 3) * DTW_N;        // diag k (write)
    float* Dm1 = s_D + ((k + 2) % 3) * DTW_N;  // diag k-1
    float* Dm2 = s_D + ((k + 1) % 3) * DTW_N;  // diag k-2

    int ilo = (k > DTW_N - 1) ? (k - (DTW_N - 1)) : 0;
    int ib  = (k >= DTW_W) ? ((k - DTW_W + 1) >> 1) : 0;   // ceil((k-20)/2)
    if (ib > ilo) ilo = ib;
    int ihi = (k < DTW_N - 1) ? k : (DTW_N - 1);
    int ub  = (k + DTW_W) >> 1;                             // floor((k+20)/2)
    if (ub < ihi) ihi = ub;

    const int i = ilo + lane;
    if (i <= ihi) {                       // <= 21 active lanes
      const int j = k - i;
      float4 p = s_pred[i];
      float4 t = s_targ[j];
      float dk = fabsf(p.x - t.x) + fabsf(p.y - t.y);
      float left = (j > 0)          ? Dm1[i]     : BIGF;   // D[i][j-1]
      float up   = (i > 0)          ? Dm1[i - 1] : BIGF;   // D[i-1][j]
      float dg   = (i > 0 && j > 0) ? Dm2[i - 1] : BIGF;   // D[i-1][j-1]
      float best = fminf(fminf(left, up), dg);
      if (k == 0) best = 0.0f;            // only cell (0,0)
      W[i] = dk + best;
      // argmin([diag, up, left]) with first-min tie-break (matches jnp.argmin)
      int c = (dg <= up && dg <= left) ? 0 : ((up <= left) ? 1 : 2);
      s_ch[i * DTW_BAND + (j - i + DTW_W)] = (unsigned char)c;
    } else if (lane == 30) {
      if (ilo > 0) W[ilo - 1] = BIGF;     // lower band-edge guard
    } else if (lane == 31) {
      if (ihi + 1 < DTW_N) W[ihi + 1] = BIGF;  // upper band-edge guard
    }
    __syncthreads();                      // single-wave WG: barrier -> S_NOP
  }

  // ---- Phase D: backtrack + loss (sequential pointer chase) ------------
  if (lane == 0) {
    const float sc0 = subcoef[0];
    const float sc1 = subcoef[1];
    int i = DTW_N - 1, j = DTW_N - 1;
    float accL = 0.0f, accB = 0.0f;
    int cnt = 0;
    while (true) {
      float4 p = s_pred[i];
      float4 t = s_targ[j];
      accL += fabsf(p.x - t.x) * sc0 + fabsf(p.y - t.y) * sc1;
      float x = fminf(fmaxf(p.z, -4.0f), 4.0f);
      float y = t.z;
      accB += 5.0f * y * softplusf(-x) + (1.0f - y) * softplusf(x);
      ++cnt;
      if (i == 0 && j == 0) break;
      int c = s_ch[i * DTW_BAND + (j - i + DTW_W)];
      int ni = (c == 2) ? i : i - 1;
      int nj = (c == 1) ? j : j - 1;
      i = ni; j = nj;
    }
    part[b] = accL + 0.1f * (accB / (float)cnt);
  }
}

// ---------------------------------------------------------------------------
// Kernel 2: deterministic 32-way sum of per-batch losses via
// V_WMMA_F32_16X16X4_F32 (A = values in documented 16x4 f32 layout, B = ones).
// ---------------------------------------------------------------------------
__global__ void __launch_bounds__(32)
finalize_kernel(const float* __restrict__ part, float* __restrict__ out)
{
#if __has_builtin(__builtin_amdgcn_wmma_f32_16x16x4_f32)
  typedef __attribute__((ext_vector_type(2))) float v2f;
  typedef __attribute__((ext_vector_type(8))) float v8f;
  const int lane = threadIdx.x;
  const int m  = lane & 15;
  const int hi = lane >> 4;   // lanes 16-31 hold K=2,3 per ISA 16x4 A layout
  v2f a;
  a.x = (m < 8) ? part[4 * m + (hi ? 2 : 0)] : 0.0f;  // VGPR0: K = hi?2:0
  a.y = (m < 8) ? part[4 * m + (hi ? 3 : 1)] : 0.0f;  // VGPR1: K = hi?3:1
  v2f bones;
  bones.x = 1.0f;
  bones.y = 1.0f;   // all-ones B: layout invariant
  v8f c = {};
  c = __builtin_amdgcn_wmma_f32_16x16x4_f32(
      /*neg_a=*/false, a, /*neg_b=*/false, bones,
      /*c_mod=*/(short)0, c, /*reuse_a=*/false, /*reuse_b=*/false);
  if (lane == 0) {
    // D[v][0] sits in VGPR v, lane 0 (rows 0..7 hold the 8 partial sums)
    out[0] = c[0] + c[1] + c[2] + c[3] + c[4] + c[5] + c[6] + c[7];
  }
#else
  if (threadIdx.x == 0) {
    float s = 0.0f;
    for (int i = 0; i < DTW_B; ++i) s += part[i];
    out[0] = s;
  }
#endif
}

// ---------------------------------------------------------------------------
extern "C" void kernel_launch(void* const* d_in, const int* in_sizes, int n_in,
                              void* d_out, int out_size, void* d_ws, size_t ws_size,
                              hipStream_t stream)
{
  (void)in_sizes; (void)n_in; (void)out_size; (void)ws_size;
  const float* preds   = (const float*)d_in[0];   // [32,1024,4] f32
  const float* targs   = (const float*)d_in[1];   // [32,1024,4] f32
  const float* subcoef = (const float*)d_in[2];   // [2] f32
  float* out  = (float*)d_out;                    // [1] f32
  float* part = (float*)d_ws;                     // 32 f32 partials

  const size_t smem = 16 * DTW_N      // s_pred
                    + 16 * DTW_N      // s_targ
                    + 12 * DTW_N      // 3 wavefront buffers
                    + (size_t)DTW_N * DTW_BAND;  // choice matrix
  dtw_kernel<<<DTW_B, 32, smem, stream>>>(preds, targs, subcoef, part);
  finalize_kernel<<<1, 32, 0, stream>>>(part, out);
}